// FSI_Initial_24919400251605
// MI455X (gfx1250) — compile-verified
//
#include <hip/hip_runtime.h>
#include <hip/hip_bf16.h>
#include <stdint.h>

// Problem constants (from reference setup_inputs)
static constexpr int B_ = 8;
static constexpr int H_ = 1024;
static constexpr int W_ = 1024;
static constexpr int C_ = 3;
static constexpr int HW_ = H_ * W_;                 // 1,048,576
static constexpr int M_ = HW_ * C_;                 // 3,145,728 elements per batch
static constexpr uint32_t R_ = (M_ + 1) / 2;        // ceil(M/P), P=2 -> 1,572,864

static constexpr int THREADS = 256;
static constexpr int WAVES = THREADS / 32;
static constexpr int BPB = 128;                     // histogram blocks per batch
static constexpr int U4_PER_BATCH = M_ / 4;         // 786,432 float4s
static constexpr int U4_PER_BLOCK = U4_PER_BATCH / BPB;   // 6144
static constexpr int U4_PER_WAVE = U4_PER_BLOCK / WAVES;  // 768
static constexpr int HIST_ITERS = U4_PER_WAVE / 32;       // 24
static constexpr int NBZ = 512;                     // Z0 reduction blocks

// ---- monotone float<->sortable-uint transform -------------------------------
__device__ __forceinline__ uint32_t fkey(float f) {
  uint32_t u = __float_as_uint(f);
  return (u & 0x80000000u) ? ~u : (u | 0x80000000u);
}
__device__ __forceinline__ float funkey(uint32_t k) {
  uint32_t u = (k & 0x80000000u) ? (k & 0x7FFFFFFFu) : ~k;
  return __uint_as_float(u);
}

// ---- CDNA5 async global->LDS copy (ASYNCcnt tracked) ------------------------
__device__ __forceinline__ void async_load_b128(uint32_t lds_off, const void* gptr) {
  asm volatile("global_load_async_to_lds_b128 %0, %1, off"
               :: "v"(lds_off), "v"((unsigned long long)(uintptr_t)gptr)
               : "memory");
}
__device__ __forceinline__ void wait_async_le1() {
  asm volatile("s_wait_asynccnt 1" ::: "memory");
}
__device__ __forceinline__ void wait_async_le0() {
  asm volatile("s_wait_asynccnt 0" ::: "memory");
}

// ---- radix-select histogram pass (levels 0/1/2), async double-buffered ------
__global__ void __launch_bounds__(THREADS)
hist_kernel(const float* __restrict__ Y, const uint32_t* __restrict__ state,
            uint32_t* __restrict__ hist, int level) {
  __shared__ uint32_t hist_s[4096];
  __shared__ float4 stage[2][WAVES][32];

  const int tid = threadIdx.x;
  const int wave = tid >> 5;
  const int lane = tid & 31;
  const int b = blockIdx.y;

  const float4* src = (const float4*)(Y + (size_t)b * (size_t)M_)
                      + (size_t)blockIdx.x * U4_PER_BLOCK + (size_t)wave * U4_PER_WAVE;

  // Prime buffer 0 via the CDNA5 async global->LDS pipe FIRST, so the HBM
  // fetch latency overlaps the 16KB LDS histogram zeroing below.
  async_load_b128((uint32_t)(uintptr_t)&stage[0][wave][lane], src + lane);

#pragma unroll
  for (int j = 0; j < 4096 / THREADS; ++j) hist_s[tid + THREADS * j] = 0;
  __syncthreads();

  const uint32_t pk = state[2 * b];  // uniform per batch (prefix key so far)

  for (int it = 0; it < HIST_ITERS; ++it) {
    const int cur = it & 1;
    if (it + 1 < HIST_ITERS) {
      async_load_b128((uint32_t)(uintptr_t)&stage[(it + 1) & 1][wave][lane],
                      src + (it + 1) * 32 + lane);
      wait_async_le1();   // oldest (current buffer) complete; async loads finish in order
    } else {
      wait_async_le0();
    }
    float4 v = stage[cur][wave][lane];
    uint32_t ks[4] = { fkey(v.x), fkey(v.y), fkey(v.z), fkey(v.w) };
#pragma unroll
    for (int j = 0; j < 4; ++j) {
      uint32_t k = ks[j];
      uint32_t bin;
      bool match;
      if (level == 0)      { match = true;                      bin = k >> 20;          }
      else if (level == 1) { match = ((k >> 20) == (pk >> 20)); bin = (k >> 8) & 0xFFFu; }
      else                 { match = ((k >> 8)  == (pk >> 8));  bin = k & 0xFFu;         }
      if (match) atomicAdd(&hist_s[bin], 1u);
    }
  }
  __syncthreads();

#pragma unroll
  for (int j = 0; j < 4096 / THREADS; ++j) {
    const int i = tid + THREADS * j;
    uint32_t c = hist_s[i];
    if (c) atomicAdd(&hist[b * 4096 + i], c);
  }
}

// ---- init: zero histograms, seed per-batch state ----------------------------
__global__ void init_kernel(uint32_t* __restrict__ hist, uint32_t* __restrict__ state) {
  int gid = blockIdx.x * blockDim.x + threadIdx.x;
  if (gid < B_ * 4096) hist[gid] = 0;
  if (gid < B_) { state[2 * gid] = 0u; state[2 * gid + 1] = R_; }
}

// ---- scan histogram descending, pick bin containing k-th largest ------------
__global__ void select_kernel(uint32_t* __restrict__ hist, uint32_t* __restrict__ state,
                              uint32_t* __restrict__ thrBits, int level) {
  const int b = blockIdx.x;
  const int nbins = (level == 2) ? 256 : 4096;
  const int shift = (level == 0) ? 20 : (level == 1 ? 8 : 0);
  if (threadIdx.x == 0) {
    const uint32_t k = state[2 * b + 1];
    uint32_t pk = state[2 * b];
    uint32_t cum = 0, krem = k;
    int sel = 0;
    for (int bin = nbins - 1; bin >= 0; --bin) {
      uint32_t c = hist[b * 4096 + bin];
      if (cum + c >= k) { sel = bin; krem = k - cum; break; }
      cum += c;
    }
    pk |= ((uint32_t)sel) << shift;
    state[2 * b] = pk;
    state[2 * b + 1] = krem;
    if (level == 2) thrBits[b] = pk;   // exact 32-bit key of R-th largest
  }
  __syncthreads();
  for (int i = threadIdx.x; i < 4096; i += blockDim.x) hist[b * 4096 + i] = 0;
}

// ---- thresholded mask, NHWC -> NCHW transpose, float4 streams ---------------
__global__ void __launch_bounds__(THREADS)
compare_kernel(const float* __restrict__ Y, const uint32_t* __restrict__ thrBits,
               float* __restrict__ out) {
  const int b = blockIdx.y;
  const int t = blockIdx.x * THREADS + threadIdx.x;   // one thread = 4 pixels
  // Y/1024 >= thr/1024  <=>  Y >= thr_raw  (exact power-of-two scaling)
  const float thr = funkey(thrBits[b]);

  const float4* src = (const float4*)(Y + (size_t)b * (size_t)M_) + 3 * (size_t)t;
  float4 a = src[0], d = src[1], e = src[2];   // 12 floats = 4 NHWC pixels

  const size_t base = (size_t)b * 3 * (size_t)HW_ + 4 * (size_t)t;
  float4 r0 = { a.x >= thr ? 1.f : 0.f, a.w >= thr ? 1.f : 0.f,
                d.z >= thr ? 1.f : 0.f, e.y >= thr ? 1.f : 0.f };   // c=0
  float4 r1 = { a.y >= thr ? 1.f : 0.f, d.x >= thr ? 1.f : 0.f,
                d.w >= thr ? 1.f : 0.f, e.z >= thr ? 1.f : 0.f };   // c=1
  float4 r2 = { a.z >= thr ? 1.f : 0.f, d.y >= thr ? 1.f : 0.f,
                e.x >= thr ? 1.f : 0.f, e.w >= thr ? 1.f : 0.f };   // c=2
  *(float4*)(out + base)                   = r0;
  *(float4*)(out + base + HW_)             = r1;
  *(float4*)(out + base + 2 * (size_t)HW_) = r2;
}

// ---- Z0 Frobenius norm: deterministic two-stage reduction -------------------
__global__ void z0_partial_kernel(const float* __restrict__ A, float* __restrict__ partial) {
  __shared__ float red[THREADS];
  const int tid = threadIdx.x;
  float s = 0.f;
  for (size_t i = (size_t)blockIdx.x * THREADS + tid; i < (size_t)HW_;
       i += (size_t)NBZ * THREADS) {
    float v = A[i];
    s += v * v;              // |Z0|^2 == Z0_abs^2 (|e^{i theta}| = 1)
  }
  red[tid] = s;
  __syncthreads();
  for (int o = THREADS / 2; o > 0; o >>= 1) {
    if (tid < o) red[tid] += red[tid + o];
    __syncthreads();
  }
  if (tid == 0) partial[blockIdx.x] = red[0];
}

__global__ void z0_final_kernel(const float* __restrict__ partial, float* __restrict__ fro) {
  __shared__ float red[THREADS];
  const int tid = threadIdx.x;
  red[tid] = partial[tid] + partial[tid + THREADS];
  __syncthreads();
  for (int o = THREADS / 2; o > 0; o >>= 1) {
    if (tid < o) red[tid] += red[tid + o];
    __syncthreads();
  }
  if (tid == 0) fro[0] = sqrtf(red[0]);
}

// ---- Z0 = |Z0| e^{i theta} / fro, interleaved re/im, float4 stores ----------
__global__ void __launch_bounds__(THREADS)
z0_out_kernel(const float* __restrict__ A, const float* __restrict__ Ang,
              const float* __restrict__ fro, float* __restrict__ out) {
  const size_t t = (size_t)blockIdx.x * THREADS + threadIdx.x;  // one thread = 2 elems
  const float f = fro[0];
  float2 a = ((const float2*)A)[t];
  float2 g = ((const float2*)Ang)[t];
  float s0, c0, s1, c1;
  __sincosf(g.x, &s0, &c0);
  __sincosf(g.y, &s1, &c1);
  float4 r = { a.x * c0 / f, a.x * s0 / f, a.y * c1 / f, a.y * s1 / f };
  ((float4*)out)[t] = r;
}

extern "C" void kernel_launch(void* const* d_in, const int* in_sizes, int n_in,
                              void* d_out, int out_size, void* d_ws, size_t ws_size,
                              hipStream_t stream) {
  (void)in_sizes; (void)n_in; (void)out_size; (void)ws_size;
  const float* Y       = (const float*)d_in[0];   // [B,H,W,C]
  const float* Z0_abs  = (const float*)d_in[1];   // [H,W]
  const float* Z0_ang  = (const float*)d_in[2];   // [H,W]
  float* out = (float*)d_out;

  // workspace layout
  uint32_t* hist    = (uint32_t*)d_ws;            // B*4096
  uint32_t* state   = hist + B_ * 4096;           // 2*B (prefix key, k-remaining)
  uint32_t* thrBits = state + 2 * B_;             // B
  float*    partial = (float*)(thrBits + B_);     // NBZ
  float*    fro     = partial + NBZ;              // 1

  init_kernel<<<(B_ * 4096) / THREADS, THREADS, 0, stream>>>(hist, state);

  for (int level = 0; level < 3; ++level) {
    hist_kernel<<<dim3(BPB, B_), THREADS, 0, stream>>>(Y, state, hist, level);
    select_kernel<<<B_, THREADS, 0, stream>>>(hist, state, thrBits, level);
  }

  compare_kernel<<<dim3(HW_ / 4 / THREADS, B_), THREADS, 0, stream>>>(Y, thrBits, out);

  z0_partial_kernel<<<NBZ, THREADS, 0, stream>>>(Z0_abs, partial);
  z0_final_kernel<<<1, THREADS, 0, stream>>>(partial, fro);
  z0_out_kernel<<<HW_ / 2 / THREADS, THREADS, 0, stream>>>(
      Z0_abs, Z0_ang, fro, out + (size_t)B_ * 3 * (size_t)HW_);
}